// GcnNet_57423712747784
// MI455X (gfx1250) — compile-verified
//
#include <hip/hip_runtime.h>
#include <hip/hip_bf16.h>

#define NN 40000
#define DD 128
#define TILES (NN / 16)   // 2500, exactly 625 blocks * 4 waves

typedef float v2f __attribute__((ext_vector_type(2)));
typedef float v8f __attribute__((ext_vector_type(8)));

// ---------------------------------------------------------------------------
// Zero the two accumulator buffers (res, ppi) -- float4 stores.
// ---------------------------------------------------------------------------
__global__ void gcn_zero_kernel(float4* __restrict__ p, unsigned n4) {
  unsigned i = blockIdx.x * blockDim.x + threadIdx.x;
  if (i < n4) p[i] = make_float4(0.f, 0.f, 0.f, 0.f);
}

// ---------------------------------------------------------------------------
// Fused dual aggregation, one wave32 per edge, 4 features per lane:
//   res[dst[e]][f] += w_self[e] * h[src[e]][f]
//   ppi[dst[e]][f] += w_ppi[e]  * h[src[e]][f]
// float4 load of the h row (coalesced 512B per wave), 8 f32 atomics per lane.
// ---------------------------------------------------------------------------
__global__ void gcn_agg_kernel(const float* __restrict__ h,
                               const int*   __restrict__ src,
                               const int*   __restrict__ dst,
                               const float* __restrict__ wself,
                               const float* __restrict__ wppi,
                               float* __restrict__ res,
                               float* __restrict__ ppi,
                               unsigned nEdges) {
  unsigned idx = blockIdx.x * blockDim.x + threadIdx.x;
  unsigned e = idx >> 5;                 // 32 lanes per edge
  if (e >= nEdges) return;
  unsigned f = (idx & 31u) * 4u;         // feature base for this lane
  int s = src[e];
  int d = dst[e];
  float a = wself[e];
  float b = wppi[e];
  const float4 v = *(const float4*)(h + (size_t)s * DD + f);
  float* rp = res + (size_t)d * DD + f;
  float* pp = ppi + (size_t)d * DD + f;
  atomicAdd(rp + 0, a * v.x); atomicAdd(rp + 1, a * v.y);
  atomicAdd(rp + 2, a * v.z); atomicAdd(rp + 3, a * v.w);
  atomicAdd(pp + 0, b * v.x); atomicAdd(pp + 1, b * v.y);
  atomicAdd(pp + 2, b * v.z); atomicAdd(pp + 3, b * v.w);
}

// ---------------------------------------------------------------------------
// h_out = relu(ppi @ W^T + b) + res   via V_WMMA_F32_16X16X4_F32.
//
// Block = 128 threads = 4 waves, each wave owns one 16-row tile and all 8
// 16-col tiles (8 x v8f accumulators).
//
// W staged in LDS, transposed pair layout:
//   sW[(k>>1)*256 + col*2 + (k&1)] = W[col][k]
// so the B fragment {W[col][k], W[col][k+1]} (k even) is one aligned
// ds_load_b64, bank = (2*col) mod 64 -> conflict-free across each 16-lane
// half (2-way between halves only).
//
// A tile (16x128) preloaded into 32 v2f registers per lane in one burst of
// independent global_load_b64s; the MAC loop then runs entirely out of
// registers + LDS.
//
// Fragment layouts (CDNA5 ISA 7.12.2, 32-bit operands):
//   A 16x4: lane%16 = M, K = (lane>>4)*2 + vgpr
//   B 4x16: lane%16 = N, K = (lane>>4)*2 + vgpr
//   C 16x16: vgpr r, lane L -> D[M = (L>>4)*8 + r][N = 16j + L%16]
// ---------------------------------------------------------------------------
__global__ void gcn_gemm_kernel(const float* __restrict__ ppi,
                                const float* __restrict__ W,     // [DD][DD]
                                const float* __restrict__ bias,  // [DD]
                                const float* __restrict__ res,
                                float* __restrict__ hout) {
  __shared__ float sW[64 * 256];         // 65536 B = 64 KB

  const int tid  = threadIdx.x;
  const int lane = tid & 31;
  const int wave = tid >> 5;
  const int tile = blockIdx.x * 4 + wave;   // grid is exact: tile < TILES always

  const int m    = lane & 15;            // M for A, N for B/C
  const int half = lane >> 4;

  // ---- Stage W into LDS (all 128 threads; 4096 float4 reads total). ----
  // idx -> (n = row of W, kq = k/4); read W[n][4kq..4kq+3] as float4,
  // write two float2s into the transposed pair layout.
#pragma unroll
  for (int it = 0; it < 32; ++it) {
    int idx = it * 128 + tid;            // 0..4095
    int n   = idx >> 5;                  // 0..127
    int k   = (idx & 31) << 2;           // 0,4,...,124
    float4 w = *(const float4*)(W + (size_t)n * DD + k);
    v2f lo = {w.x, w.y};
    v2f hi = {w.z, w.w};
    *(v2f*)(sW + ((k >> 1) + 0) * 256 + n * 2) = lo;
    *(v2f*)(sW + ((k >> 1) + 1) * 256 + n * 2) = hi;
  }
  __syncthreads();

  // ---- Preload the whole A tile row for this lane (32 x b64, one burst). --
  const float* aRow = ppi + (size_t)(tile * 16 + m) * DD + 2 * half;
  v2f aReg[32];
#pragma unroll
  for (int i = 0; i < 32; ++i)
    aReg[i] = *(const v2f*)(aRow + i * 4);

  // ---- Init accumulators with bias (bias varies only along N=col). -------
  v8f c[8];
#pragma unroll
  for (int j = 0; j < 8; ++j) {
    float bv = bias[j * 16 + m];
#pragma unroll
    for (int r = 0; r < 8; ++r) c[j][r] = bv;
  }

  // ---- MAC loop: K in steps of 4; 8 ds_load_b64 + 8 WMMA per step. -------
  const float* bBase = sW + half * 256 + m * 2;
#pragma unroll
  for (int kb2 = 0; kb2 < 64; kb2 += 2) {   // kb2 = kb/2
    const float* bp = bBase + kb2 * 256;
    v2f bf[8];
#pragma unroll
    for (int j = 0; j < 8; ++j) bf[j] = *(const v2f*)(bp + j * 32);
    v2f a = aReg[kb2 >> 1];
#pragma unroll
    for (int j = 0; j < 8; ++j) {
      c[j] = __builtin_amdgcn_wmma_f32_16x16x4_f32(
          /*neg_a=*/false, a, /*neg_b=*/false, bf[j],
          /*c_mod=*/(short)0, c[j], /*reuse_a=*/false, /*reuse_b=*/false);
    }
  }

  // ---- Epilogue: relu + residual add, scatter per C/D layout. ------------
#pragma unroll
  for (int j = 0; j < 8; ++j) {
    const int col = j * 16 + m;
#pragma unroll
    for (int r = 0; r < 8; ++r) {
      const int row = tile * 16 + half * 8 + r;
      float v = c[j][r];
      v = v > 0.0f ? v : 0.0f;
      hout[(size_t)row * DD + col] = v + res[(size_t)row * DD + col];
    }
  }
}

// ---------------------------------------------------------------------------
// Launch: per layer l in {0,1}:
//   zero(res,ppi) -> agg(h_l) -> gemm(ppi, W[l], b[l], res) -> h_{l+1}
// h0 = d_in[0]; h1 = d_out (overwritten by final layer); res/ppi in d_ws.
// ---------------------------------------------------------------------------
extern "C" void kernel_launch(void* const* d_in, const int* in_sizes, int n_in,
                              void* d_out, int out_size, void* d_ws, size_t ws_size,
                              hipStream_t stream) {
  const float* h0    = (const float*)d_in[0];  // [N, D]
  const int*   esrc  = (const int*)  d_in[1];  // [E]
  const int*   edst  = (const int*)  d_in[2];  // [E]
  const float* wself = (const float*)d_in[3];  // [E]
  const float* wppi  = (const float*)d_in[4];  // [E]
  const float* Wmat  = (const float*)d_in[5];  // [L, D, D]
  const float* bvec  = (const float*)d_in[6];  // [L, D]
  float* out = (float*)d_out;

  const unsigned E = (unsigned)in_sizes[1];
  float* res = (float*)d_ws;
  float* ppi = res + (size_t)NN * DD;

  const unsigned n4       = (2u * NN * DD) / 4u;
  const unsigned zeroGrid = (n4 + 255u) / 256u;
  const unsigned aggThr   = E * 32u;                 // one wave per edge
  const unsigned aggGrid  = (aggThr + 255u) / 256u;
  const unsigned gemmGrid = TILES / 4;               // 625, exact

  for (int l = 0; l < 2; ++l) {
    const float* hin = (l == 0) ? h0 : out;
    gcn_zero_kernel<<<zeroGrid, 256, 0, stream>>>((float4*)res, n4);
    gcn_agg_kernel<<<aggGrid, 256, 0, stream>>>(hin, esrc, edst, wself, wppi,
                                                res, ppi, E);
    gcn_gemm_kernel<<<gemmGrid, 128, 0, stream>>>(ppi,
                                                  Wmat + (size_t)l * DD * DD,
                                                  bvec + (size_t)l * DD,
                                                  res, out);
  }
}